// HeteroGraphSage_49512382988850
// MI455X (gfx1250) — compile-verified
//
#include <hip/hip_runtime.h>
#include <math.h>

// ---------------------------------------------------------------------------
// HeteroGraphSage for MI455X (gfx1250, wave32).
// Dense projections via V_WMMA_F32_16X16X4_F32 (fp32-exact path, the compute
// volume ~49 GFLOP is negligible next to the ~1.5GB of edge traffic, so we
// keep fp32 fidelity for the LayerNorm-heavy reference).
// Segment-mean aggregation via global float atomics (L2-resident, 192MB L2).
// ---------------------------------------------------------------------------

typedef __attribute__((ext_vector_type(2))) float v2f;
typedef __attribute__((ext_vector_type(8))) float v8f;

#define H      128
#define NOPN   100000
#define NRTN   100000
#define NEDGE  200000
#define DOP    256
#define GEMM_GRID 512

// -------------------------- wave32 reduction -------------------------------
__device__ __forceinline__ float wave_sum32(float v) {
#pragma unroll
  for (int off = 16; off > 0; off >>= 1)
    v += __shfl_xor(v, off, 32);
  return v;
}

// ---------------------------------------------------------------------------
// GEMM: out[N,128]  (=|+=)  scale(A)[N,K] @ W[K,128]  (+ bias)
//   scale(A) row r = A[r] * 1/max(deg[r],1)  when DEG (fused mean-agg)
// Block = 256 threads = 8 waves; wave w owns columns 16w..16w+15 and caches
// its whole B panel in registers (v2f[K/4]); loops over 16-row strips.
// WMMA f32 16x16x4 fragment layout per ISA 7.12.2:
//   A: lane L -> row (L&15), k = 4*kt + 2*(L>>4) + {0,1}
//   B: lane L -> col (L&15), k = 4*kt + 2*(L>>4) + {0,1}
//   C: vgpr r -> row r + 8*(L>>4), col (L&15)
// ---------------------------------------------------------------------------
template <int K, bool ACC, bool BIAS, bool DEG>
__global__ __launch_bounds__(256)
void gemm_wmma_f32(const float* __restrict__ A,
                   const float* __restrict__ deg,
                   const float* __restrict__ W,
                   const float* __restrict__ bias,
                   float* __restrict__ out, int N) {
  const int lane  = threadIdx.x & 31;
  const int wave  = threadIdx.x >> 5;
  const int n     = (wave << 4) + (lane & 15);   // output column 0..127
  const int khalf = (lane >> 4) << 1;            // 0 or 2

  // Register-cache the B panel for this wave's 16 columns.
  v2f bfrag[K / 4];
#pragma unroll
  for (int kt = 0; kt < K / 4; ++kt) {
    const int kb = (kt << 2) + khalf;
    bfrag[kt].x = W[(size_t)kb * H + n];
    bfrag[kt].y = W[(size_t)(kb + 1) * H + n];
  }
  const float bval = BIAS ? bias[n] : 0.0f;

  for (int m0 = blockIdx.x * 16; m0 < N; m0 += gridDim.x * 16) {
    const int row = m0 + (lane & 15);
    float scale = 1.0f;
    if (DEG) scale = 1.0f / fmaxf(deg[row], 1.0f);
    const float* __restrict__ arow = A + (size_t)row * K;

    v8f acc = {};
#pragma unroll
    for (int kt = 0; kt < K / 4; ++kt) {
      const int kb = (kt << 2) + khalf;
      v2f a;
      a.x = arow[kb] * scale;
      a.y = arow[kb + 1] * scale;
      acc = __builtin_amdgcn_wmma_f32_16x16x4_f32(
          false, a, false, bfrag[kt], (short)0, acc, false, false);
    }

    const int rbase = m0 + ((lane >> 4) << 3);
#pragma unroll
    for (int r = 0; r < 8; ++r) {
      const size_t o = (size_t)(rbase + r) * H + n;
      float v = acc[r] + bval;
      if (ACC) v += out[o];
      out[o] = v;
    }
  }
}

// ---------------------------------------------------------------------------
// Edge scatter: agg[dst] += h_src[src] (per feature), deg[dst] += 1.
// One thread per (edge, feature); atomics resolve in L2.
// ---------------------------------------------------------------------------
__global__ __launch_bounds__(256)
void scatter_edges(const float* __restrict__ hsrc,
                   const int* __restrict__ src,
                   const int* __restrict__ dst,
                   float* __restrict__ agg,
                   float* __restrict__ deg, int E) {
  const long long gid = (long long)blockIdx.x * blockDim.x + threadIdx.x;
  const int e = (int)(gid >> 7);
  const int f = (int)(gid & (H - 1));
  if (e >= E) return;
  const int s = src[e];
  const int d = dst[e];
  atomicAdd(&agg[(size_t)d * H + f], hsrc[(size_t)s * H + f]);
  if (f == 0) atomicAdd(&deg[d], 1.0f);
}

// ---------------------------------------------------------------------------
// Row-wise LayerNorm + exact GELU. One wave per 128-wide row (4 elts/lane).
// ---------------------------------------------------------------------------
__global__ __launch_bounds__(256)
void ln_gelu(const float* __restrict__ x,
             const float* __restrict__ g,
             const float* __restrict__ b,
             float* __restrict__ out, int N) {
  const int lane = threadIdx.x & 31;
  const int wave = threadIdx.x >> 5;
  const int row  = blockIdx.x * 8 + wave;
  if (row >= N) return;
  const float* __restrict__ xr = x + (size_t)row * H;

  float v[4];
  float s = 0.0f;
#pragma unroll
  for (int i = 0; i < 4; ++i) { v[i] = xr[lane + i * 32]; s += v[i]; }
  const float mu = wave_sum32(s) * (1.0f / H);

  float var = 0.0f;
#pragma unroll
  for (int i = 0; i < 4; ++i) { const float d = v[i] - mu; var += d * d; }
  var = wave_sum32(var) * (1.0f / H);
  const float rstd = rsqrtf(var + 1e-5f);

#pragma unroll
  for (int i = 0; i < 4; ++i) {
    const int c = lane + i * 32;
    const float xn = (v[i] - mu) * rstd * g[c] + b[c];
    out[(size_t)row * H + c] = 0.5f * xn * (1.0f + erff(xn * 0.70710678118654752f));
  }
}

// out[i] = h_rt[i] . w + b     (one wave per row)
__global__ __launch_bounds__(256)
void out_proj(const float* __restrict__ hrt,
              const float* __restrict__ w,
              const float* __restrict__ b,
              float* __restrict__ out, int N) {
  const int lane = threadIdx.x & 31;
  const int wave = threadIdx.x >> 5;
  const int row  = blockIdx.x * 8 + wave;
  if (row >= N) return;
  float s = 0.0f;
#pragma unroll
  for (int i = 0; i < 4; ++i) {
    const int c = lane + i * 32;
    s += hrt[(size_t)row * H + c] * w[c];
  }
  s = wave_sum32(s);
  if (lane == 0) out[row] = s + b[0];
}

// c = a + b (elementwise) — combine w_self / bias of the two op-dst relations
__global__ void add_vec(const float* __restrict__ a, const float* __restrict__ b,
                        float* __restrict__ c, int n) {
  const int i = blockIdx.x * blockDim.x + threadIdx.x;
  if (i < n) c[i] = a[i] + b[i];
}

// ---------------------------------------------------------------------------
// Input order (jax tree-leaf / sorted-key flattening of setup_inputs()):
//  0 feat_op [1e5,256]   1 feat_rt [1e5,128]
//  adapt.op:  2 b, 3 beta, 4 g, 5 w[256,128]
//  adapt.rt:  6 b, 7 beta, 8 g, 9 w[128,128]
//  layer l (base 10+11l): ln_b, ln_g, r0.bias, r0.w_neigh, r0.w_self,
//                         r1.bias, r1.w_neigh, r1.w_self,
//                         r2.bias, r2.w_neigh, r2.w_self
//  32 out.b [1], 33 out.w [128,1]
//  34 r0_src 35 r0_dst 36 r1_src 37 r1_dst 38 r2_src 39 r2_dst
// ---------------------------------------------------------------------------
extern "C" void kernel_launch(void* const* d_in, const int* in_sizes, int n_in,
                              void* d_out, int out_size, void* d_ws, size_t ws_size,
                              hipStream_t stream) {
  const float* feat_op = (const float*)d_in[0];
  const float* feat_rt = (const float*)d_in[1];
  const float* aop_b    = (const float*)d_in[2];
  const float* aop_beta = (const float*)d_in[3];
  const float* aop_g    = (const float*)d_in[4];
  const float* aop_w    = (const float*)d_in[5];
  const float* art_b    = (const float*)d_in[6];
  const float* art_beta = (const float*)d_in[7];
  const float* art_g    = (const float*)d_in[8];
  const float* art_w    = (const float*)d_in[9];
  const float* out_b = (const float*)d_in[32];
  const float* out_w = (const float*)d_in[33];
  const int* r0_src = (const int*)d_in[34];
  const int* r0_dst = (const int*)d_in[35];
  const int* r1_src = (const int*)d_in[36];
  const int* r1_dst = (const int*)d_in[37];
  const int* r2_src = (const int*)d_in[38];
  const int* r2_dst = (const int*)d_in[39];

  // workspace carve-up (~256.6 MB)
  float* ws     = (float*)d_ws;
  float* h_op   = ws;                    ws += (size_t)NOPN * H;
  float* h_rt   = ws;                    ws += (size_t)NRTN * H;
  float* new_op = ws;                    ws += (size_t)NOPN * H;
  float* new_rt = ws;                    ws += (size_t)NRTN * H;
  float* agg    = ws;                    ws += (size_t)NOPN * H;
  float* deg    = ws;                    ws += (size_t)NOPN;
  float* wc     = ws;                    ws += (size_t)H * H;
  float* bc     = ws;                    ws += (size_t)H;

  const dim3 blk(256);
  const int rowsOp = NOPN / 8, rowsRt = NRTN / 8;       // 12500 each
  const int scatGrid = (int)(((long long)NEDGE * H) / 256);

  // ---- adapt + LN + GELU ----
  gemm_wmma_f32<DOP, false, true, false><<<GEMM_GRID, blk, 0, stream>>>(
      feat_op, nullptr, aop_w, aop_b, new_op, NOPN);
  ln_gelu<<<rowsOp, blk, 0, stream>>>(new_op, aop_g, aop_beta, h_op, NOPN);
  gemm_wmma_f32<H, false, true, false><<<GEMM_GRID, blk, 0, stream>>>(
      feat_rt, nullptr, art_w, art_b, new_rt, NRTN);
  ln_gelu<<<rowsRt, blk, 0, stream>>>(new_rt, art_g, art_beta, h_rt, NRTN);

  // ---- 2 hetero-SAGE layers ----
  for (int l = 0; l < 2; ++l) {
    const int base = 10 + 11 * l;
    const float* ln_b     = (const float*)d_in[base + 0];
    const float* ln_g     = (const float*)d_in[base + 1];
    const float* r0_bias  = (const float*)d_in[base + 2];
    const float* r0_wn    = (const float*)d_in[base + 3];
    const float* r0_ws    = (const float*)d_in[base + 4];
    const float* r1_bias  = (const float*)d_in[base + 5];
    const float* r1_wn    = (const float*)d_in[base + 6];
    const float* r1_ws    = (const float*)d_in[base + 7];
    const float* r2_bias  = (const float*)d_in[base + 8];
    const float* r2_wn    = (const float*)d_in[base + 9];
    const float* r2_ws_   = (const float*)d_in[base + 10];

    // combined self path for op dst: h_op @ (W0s + W2s) + (b0 + b2)
    add_vec<<<(H * H + 255) / 256, blk, 0, stream>>>(r0_ws, r2_ws_, wc, H * H);
    add_vec<<<1, blk, 0, stream>>>(r0_bias, r2_bias, bc, H);
    gemm_wmma_f32<H, false, true, false><<<GEMM_GRID, blk, 0, stream>>>(
        h_op, nullptr, wc, bc, new_op, NOPN);

    // r0: op -> op neighbor mean
    hipMemsetAsync(agg, 0, (size_t)NOPN * H * sizeof(float), stream);
    hipMemsetAsync(deg, 0, (size_t)NOPN * sizeof(float), stream);
    scatter_edges<<<scatGrid, blk, 0, stream>>>(h_op, r0_src, r0_dst, agg, deg, NEDGE);
    gemm_wmma_f32<H, true, false, true><<<GEMM_GRID, blk, 0, stream>>>(
        agg, deg, r0_wn, nullptr, new_op, NOPN);

    // r2: rt -> op neighbor mean
    hipMemsetAsync(agg, 0, (size_t)NOPN * H * sizeof(float), stream);
    hipMemsetAsync(deg, 0, (size_t)NOPN * sizeof(float), stream);
    scatter_edges<<<scatGrid, blk, 0, stream>>>(h_rt, r2_src, r2_dst, agg, deg, NEDGE);
    gemm_wmma_f32<H, true, false, true><<<GEMM_GRID, blk, 0, stream>>>(
        agg, deg, r2_wn, nullptr, new_op, NOPN);

    // r1: op -> rt (self + neighbor mean)
    gemm_wmma_f32<H, false, true, false><<<GEMM_GRID, blk, 0, stream>>>(
        h_rt, nullptr, r1_ws, r1_bias, new_rt, NRTN);
    hipMemsetAsync(agg, 0, (size_t)NRTN * H * sizeof(float), stream);
    hipMemsetAsync(deg, 0, (size_t)NRTN * sizeof(float), stream);
    scatter_edges<<<scatGrid, blk, 0, stream>>>(h_op, r1_src, r1_dst, agg, deg, NEDGE);
    gemm_wmma_f32<H, true, false, true><<<GEMM_GRID, blk, 0, stream>>>(
        agg, deg, r1_wn, nullptr, new_rt, NRTN);

    // LN + GELU into next-iteration features
    ln_gelu<<<rowsOp, blk, 0, stream>>>(new_op, ln_g, ln_b, h_op, NOPN);
    ln_gelu<<<rowsRt, blk, 0, stream>>>(new_rt, ln_g, ln_b, h_rt, NRTN);
  }

  // ---- output projection ----
  out_proj<<<rowsRt, blk, 0, stream>>>(h_rt, out_w, out_b, (float*)d_out, NRTN);

  (void)in_sizes; (void)n_in; (void)out_size; (void)ws_size;
}